// RTDETRPostProcessor_17325898072217
// MI455X (gfx1250) — compile-verified
//
#include <hip/hip_runtime.h>
#include <stdint.h>

// ---------------- problem constants (from reference) ----------------
#define NUM_B   128
#define NQ      8192
#define NC      36
#define NFLAT   (NQ * NC)            // 294912 scores per image
#define TOPK    300
#define NBINS   4096                 // 12-bit radix histogram
#define CAP     2048                 // per-batch candidate capacity

// ---------------- streaming tiling ----------------
#define CHUNK            1024                      // floats per TDM transfer (4 KB)
#define WAVES            8                         // 256 threads = 8 wave32
#define CHUNKS_PER_WAVE  4
#define WAVE_SPAN        (CHUNKS_PER_WAVE * CHUNK) // 4096 floats
#define BLOCK_SPAN       (WAVES * WAVE_SPAN)       // 32768 floats per block
#define BPB              (NFLAT / BLOCK_SPAN)      // 9 blocks per batch (exact)

typedef unsigned int u32x4 __attribute__((ext_vector_type(4)));
typedef int          i32x4 __attribute__((ext_vector_type(4)));
typedef int          i32x8 __attribute__((ext_vector_type(8)));

// Order-preserving key: larger float  <=>  larger unsigned key.
__device__ __forceinline__ uint32_t fkey(float x) {
  uint32_t u = __float_as_uint(x);
  return (u & 0x80000000u) ? ~u : (u | 0x80000000u);
}

// LDS byte offset = low 32 bits of the flat (generic) address of a __shared__
// object (ISA: LDS aperture maps flat addr[31:0] -> LDS_ADDR).
__device__ __forceinline__ unsigned lds_lo32(const void* p) {
  return (unsigned)(uintptr_t)p;
}

// Issue one TDM load of CHUNK contiguous f32 elements: global -> LDS.
// D# built per CDNA5 ISA §8.3/8.4: group0 {count=1, lds_addr, global_addr,
// type=2}, group1 {data_size=4B, tensor_dim0=CHUNK, tile_dim0=CHUNK, 1-D}.
__device__ __forceinline__ void tdm_load_row(const float* gsrc, unsigned ldsDst) {
  unsigned long long ga = (unsigned long long)(uintptr_t)gsrc;
  u32x4 g0;
  g0[0] = 1u;                                   // count=1 valid descriptor
  g0[1] = ldsDst;                               // lds_addr (bytes)
  g0[2] = (unsigned)ga;                         // global_addr[31:0]
  g0[3] = (unsigned)((ga >> 32) & 0x01FFFFFFull) | (2u << 30); // addr[56:32] | type=2
  i32x8 g1;
  g1[0] = (2 << 16);                            // workgroup_mask=0, data_size=2 (4B)
  g1[1] = (CHUNK & 0xFFFF) << 16;               // tensor_dim0[15:0] @ bits 63:48
  g1[2] = ((CHUNK >> 16) & 0xFFFF) | (1 << 16); // tensor_dim0[31:16], tensor_dim1=1
  g1[3] = (CHUNK & 0xFFFF) << 16;               // tile_dim0 @ bits 127:112
  g1[4] = 0;                                    // tile_dim1=0 (unused), tile_dim2=0
  g1[5] = CHUNK;                                // tensor_dim0_stride[31:0]
  g1[6] = 0;
  g1[7] = 0;
  i32x4 z4 = {0, 0, 0, 0};
#if __clang_major__ >= 23
  i32x8 z8 = {0, 0, 0, 0, 0, 0, 0, 0};
  __builtin_amdgcn_tensor_load_to_lds(g0, g1, z4, z4, z8, 0);
#else
  __builtin_amdgcn_tensor_load_to_lds(g0, g1, z4, z4, 0);
#endif
}

// ---------------- kernel 0: zero histogram / counters ----------------
__global__ void k_zero(uint32_t* __restrict__ p, int n) {
  int i = blockIdx.x * blockDim.x + threadIdx.x;
  if (i < n) p[i] = 0u;
}

// ---------------- kernel 1: per-batch radix histogram (TDM-staged) ----------
__global__ __launch_bounds__(256) void k_hist(const float* __restrict__ logits,
                                              uint32_t* __restrict__ gHist) {
  __shared__ uint32_t shHist[NBINS];                       // 16 KB
  __shared__ __align__(16) float stage[WAVES][2][CHUNK];   // 64 KB double buffer
  const int b   = blockIdx.y;
  const int tid = threadIdx.x;
  for (int i = tid; i < NBINS; i += 256) shHist[i] = 0u;
  __syncthreads();

  const int wave = tid >> 5;
  const int lane = tid & 31;
  const float* src = logits + (size_t)b * NFLAT +
                     (size_t)blockIdx.x * BLOCK_SPAN + (size_t)wave * WAVE_SPAN;

  // software pipeline: TDM chunk c+1 in flight while binning chunk c
  tdm_load_row(src, lds_lo32(&stage[wave][0][0]));
  #pragma unroll
  for (int c = 0; c < CHUNKS_PER_WAVE; ++c) {
    if (c + 1 < CHUNKS_PER_WAVE) {
      tdm_load_row(src + (c + 1) * CHUNK, lds_lo32(&stage[wave][(c + 1) & 1][0]));
      __builtin_amdgcn_s_wait_tensorcnt(1);   // chunk c complete (in-order)
    } else {
      __builtin_amdgcn_s_wait_tensorcnt(0);
    }
    const float4* buf = (const float4*)&stage[wave][c & 1][0];
    #pragma unroll
    for (int k = 0; k < CHUNK / 128; ++k) {   // 32 lanes * 8 * float4 = 1024
      float4 v = buf[lane + k * 32];
      atomicAdd(&shHist[fkey(v.x) >> 20], 1u);
      atomicAdd(&shHist[fkey(v.y) >> 20], 1u);
      atomicAdd(&shHist[fkey(v.z) >> 20], 1u);
      atomicAdd(&shHist[fkey(v.w) >> 20], 1u);
    }
  }
  __syncthreads();
  uint32_t* h = gHist + (size_t)b * NBINS;
  for (int i = tid; i < NBINS; i += 256) {
    uint32_t c = shHist[i];
    if (c) atomicAdd(&h[i], c);
  }
}

// ---------------- kernel 2: find threshold bin (one block per batch) -------
__global__ __launch_bounds__(256) void k_thresh(const uint32_t* __restrict__ hist,
                                                uint32_t* __restrict__ thrBin) {
  __shared__ uint32_t gsum[256];
  const int b = blockIdx.x;
  const int t = threadIdx.x;
  const uint32_t* h = hist + (size_t)b * NBINS;
  // group t (counted from the TOP of the key range) covers bins
  // [NBINS-16*(t+1), NBINS-16*t)
  const int base = NBINS - 16 * (t + 1);
  uint32_t s = 0;
  #pragma unroll
  for (int j = 0; j < 16; ++j) s += h[base + j];
  gsum[t] = s;
  __syncthreads();
  if (t == 0) {
    uint32_t cum = 0;
    for (int g = 0; g < 256; ++g) {
      if (cum + gsum[g] >= TOPK) {
        const int gb = NBINS - 16 * g - 1;    // top bin of group g
        int bin = gb - 15;
        for (int j = 0; j < 16; ++j) {
          cum += h[gb - j];
          if (cum >= TOPK) { bin = gb - j; break; }
        }
        thrBin[b] = (uint32_t)bin;
        return;
      }
      cum += gsum[g];
    }
    thrBin[b] = 0u;                           // degenerate: take everything
  }
}

// ---------------- kernel 3: collect candidates (TDM-staged) ----------------
__global__ __launch_bounds__(256) void k_collect(const float* __restrict__ logits,
                                                 const uint32_t* __restrict__ thrBin,
                                                 uint32_t* __restrict__ cnt,
                                                 unsigned long long* __restrict__ cand) {
  __shared__ __align__(16) float stage[WAVES][2][CHUNK];   // 64 KB double buffer
  const int b   = blockIdx.y;
  const int tid = threadIdx.x;
  const uint32_t tb = thrBin[b];
  const int wave = tid >> 5;
  const int lane = tid & 31;
  const uint32_t waveIdx0 = (uint32_t)(blockIdx.x * BLOCK_SPAN + wave * WAVE_SPAN);
  const float* src = logits + (size_t)b * NFLAT + waveIdx0;

  tdm_load_row(src, lds_lo32(&stage[wave][0][0]));
  #pragma unroll
  for (int c = 0; c < CHUNKS_PER_WAVE; ++c) {
    if (c + 1 < CHUNKS_PER_WAVE) {
      tdm_load_row(src + (c + 1) * CHUNK, lds_lo32(&stage[wave][(c + 1) & 1][0]));
      __builtin_amdgcn_s_wait_tensorcnt(1);
    } else {
      __builtin_amdgcn_s_wait_tensorcnt(0);
    }
    const float4* buf = (const float4*)&stage[wave][c & 1][0];
    const uint32_t cIdx0 = waveIdx0 + (uint32_t)c * CHUNK;
    #pragma unroll
    for (int k = 0; k < CHUNK / 128; ++k) {
      float4 v = buf[lane + k * 32];
      const uint32_t e0 = cIdx0 + (uint32_t)(lane + k * 32) * 4u;
      float vals[4] = {v.x, v.y, v.z, v.w};
      #pragma unroll
      for (int j = 0; j < 4; ++j) {
        uint32_t key = fkey(vals[j]);
        if ((key >> 20) >= tb) {
          uint32_t pos = atomicAdd(&cnt[b], 1u);
          if (pos < CAP)
            cand[(size_t)b * CAP + pos] =
                ((unsigned long long)key << 32) | (uint32_t)~(e0 + (uint32_t)j);
        }
      }
    }
  }
}

// ---------------- kernel 4: per-batch bitonic sort + emit outputs ----------
__global__ __launch_bounds__(1024) void k_final(const float* __restrict__ quads,
                                                const float* __restrict__ sizes,
                                                const uint32_t* __restrict__ cnt,
                                                const unsigned long long* __restrict__ cand,
                                                float* __restrict__ out) {
  __shared__ unsigned long long sh[CAP];        // 16 KB
  const int b   = blockIdx.x;
  const int tid = threadIdx.x;
  // warm GL2 for the candidate block while cnt[b] is in flight
  for (int i = tid; i < CAP; i += 1024)
    __builtin_prefetch(&cand[(size_t)b * CAP + i], 0, 0);
  uint32_t n = cnt[b];
  if (n > CAP) n = CAP;
  for (int i = tid; i < CAP; i += 1024)
    sh[i] = (i < (int)n) ? cand[(size_t)b * CAP + i] : 0ull; // 0 sorts last
  __syncthreads();

  // bitonic sort, descending: key desc, then ~idx desc => index asc (ties)
  for (int k = 2; k <= CAP; k <<= 1) {
    for (int j = k >> 1; j > 0; j >>= 1) {
      for (int t = tid; t < CAP; t += 1024) {
        int ixj = t ^ j;
        if (ixj > t) {
          unsigned long long a = sh[t], c = sh[ixj];
          bool desc = ((t & k) == 0);
          if (desc ? (a < c) : (a > c)) { sh[t] = c; sh[ixj] = a; }
        }
      }
      __syncthreads();
    }
  }

  if (tid < TOPK) {
    unsigned long long pk = sh[tid];
    uint32_t key = (uint32_t)(pk >> 32);
    uint32_t idx = ~(uint32_t)(pk & 0xFFFFFFFFull);
    uint32_t lab = idx % NC;
    uint32_t q   = idx / NC;
    // recover logit from order-preserving key, apply sigmoid to top-K only
    uint32_t u = (key & 0x80000000u) ? (key ^ 0x80000000u) : ~key;
    float x = __uint_as_float(u);
    float score = 1.0f / (1.0f + __expf(-x));
    float h = sizes[b * 2 + 0];
    float w = sizes[b * 2 + 1];

    float* labOut   = out;                                   // [B, K]
    float* quadOut  = out + (size_t)NUM_B * TOPK;            // [B, K, 8]
    float* scoreOut = out + (size_t)NUM_B * TOPK * 9;        // [B, K]

    labOut[(size_t)b * TOPK + tid]   = (float)lab;
    scoreOut[(size_t)b * TOPK + tid] = score;
    const float* qp = quads + ((size_t)b * NQ + q) * 8;
    #pragma unroll
    for (int j = 0; j < 8; ++j)
      quadOut[((size_t)b * TOPK + tid) * 8 + j] = qp[j] * ((j & 1) ? w : h);
  }
}

// ---------------- host-side launch ----------------
extern "C" void kernel_launch(void* const* d_in, const int* in_sizes, int n_in,
                              void* d_out, int out_size, void* d_ws, size_t ws_size,
                              hipStream_t stream) {
  const float* logits = (const float*)d_in[0];   // [128, 8192, 36]
  const float* quads  = (const float*)d_in[1];   // [128, 8192, 8]
  const float* sizes  = (const float*)d_in[2];   // [128, 2]
  float* out = (float*)d_out;                    // labels | quads | scores (flat)

  // workspace layout (all 4B-aligned; cand region lands on 8B boundary)
  uint32_t* hist   = (uint32_t*)d_ws;                 // 128 * 4096
  uint32_t* thrBin = hist + (size_t)NUM_B * NBINS;    // 128
  uint32_t* cnt    = thrBin + NUM_B;                  // 128
  unsigned long long* cand = (unsigned long long*)(cnt + NUM_B); // 128 * 2048 u64

  const int nz = NUM_B * NBINS + 2 * NUM_B;
  k_zero<<<(nz + 255) / 256, 256, 0, stream>>>(hist, nz);
  k_hist<<<dim3(BPB, NUM_B), 256, 0, stream>>>(logits, hist);
  k_thresh<<<NUM_B, 256, 0, stream>>>(hist, thrBin);
  k_collect<<<dim3(BPB, NUM_B), 256, 0, stream>>>(logits, thrBin, cnt, cand);
  k_final<<<NUM_B, 1024, 0, stream>>>(quads, sizes, cnt, cand, out);
}